// RSCA_72353019068800
// MI455X (gfx1250) — compile-verified
//
#include <hip/hip_runtime.h>

// ---------------------------------------------------------------------------
// Problem constants (from reference): B=128, N=8192, H*W=49
// out[i, 0:8192]    = xp + self_x + cross_x
// out[i, 8192:16384]= y  + self_y + cross_y
// ---------------------------------------------------------------------------
#define BDIM 128
#define NDIM 8192
#define HW   49
#define NT   16       // channels per attention workgroup

typedef __attribute__((ext_vector_type(2))) float v2f;
typedef __attribute__((ext_vector_type(8))) float v8f;

// ---------------------------------------------------------------------------
// Kernel 1: AdaptiveAvgPool2d((1,1)) == mean over 49 contiguous floats/row.
// One block = 128 rows (24.5 KB). The block's global range is copied verbatim
// into LDS with GLOBAL_LOAD_ASYNC_TO_LDS_B128 (ASYNCcnt DMA path, no VGPR
// round-trip), then each wave reduces 16 rows via 12x v_wmma_f32_16x16x4_f32
// (K=0..47) against an all-ones B; column 48 is added scalar afterwards.
// D[i,j] = rowsum(i) replicated over j; column 0 extracted via LDS.
// Stride-49 (odd) row base => conflict-free LDS banks for A-fragment reads.
// ---------------------------------------------------------------------------
__global__ __launch_bounds__(256) void pool_wmma_kernel(const float* __restrict__ x,
                                                        float* __restrict__ xp) {
  __shared__ float tile[128 * HW];   // contiguous image of this block's floats
  __shared__ float sums[128];
  const int tid = threadIdx.x;
  const long long rowBase = (long long)blockIdx.x * 128;
  const float* gsrc = x + rowBase * HW;

  // ---- async DMA: global -> LDS, 16 B per lane per op (ASYNCcnt) ----------
  // Generic LDS pointer's low 32 bits are the LDS byte offset (ISA 10.2).
  const unsigned ldsbase = (unsigned)(size_t)&tile[0];
  for (int idx = tid; idx < (128 * HW) / 4; idx += 256) {   // 1568 float4s
    unsigned loff = ldsbase + (unsigned)idx * 16u;
    unsigned long long gaddr = (unsigned long long)(size_t)(gsrc + idx * 4);
    asm volatile("global_load_async_to_lds_b128 %0, %1, off"
                 :: "v"(loff), "v"(gaddr)
                 : "memory");
  }
  asm volatile("s_wait_asynccnt 0x0" ::: "memory");  // our wave's DMAs landed
  __syncthreads();                                   // everyone's DMAs landed

  // ---- WMMA row-sum: A-fragment (16x4 f32) ---------------------------------
  // lanes 0-15 hold M=0..15, K={0,1}; lanes 16-31 hold M=0..15, K={2,3}
  const int lane = tid & 31;           // wave32
  const int wave = tid >> 5;           // 8 waves x 16 rows
  const int rowl = wave * 16 + (lane & 15);
  const int kp   = (lane >> 4) << 1;

  v2f bones; bones[0] = 1.0f; bones[1] = 1.0f;   // B = 4x16 ones
  v8f acc = {};
#pragma unroll
  for (int kb = 0; kb < 12; ++kb) {    // K = 0..47
    v2f a;
    a[0] = tile[rowl * HW + kb * 4 + kp];
    a[1] = tile[rowl * HW + kb * 4 + kp + 1];
    // 8-arg form: (neg_a, A, neg_b, B, c_mod, C, reuse_a, reuse_b)
    acc = __builtin_amdgcn_wmma_f32_16x16x4_f32(false, a, false, bones,
                                                (short)0, acc, false, false);
  }
  // D layout: VGPR r -> M=r (lanes 0-15) / M=r+8 (lanes 16-31); take N=0 col.
  if (lane == 0) {
#pragma unroll
    for (int r = 0; r < 8; ++r) sums[wave * 16 + r] = acc[r];
  } else if (lane == 16) {
#pragma unroll
    for (int r = 0; r < 8; ++r) sums[wave * 16 + 8 + r] = acc[r];
  }
  __syncthreads();
  if (tid < 128)                        // add the K=48 tail column, scale
    xp[rowBase + tid] = (sums[tid] + tile[tid * HW + 48]) * (1.0f / 49.0f);
}

// ---------------------------------------------------------------------------
// Kernel 2: four scalar-head attentions per channel, fused.
// scores factorize (q_i * k_j), so row-max = q_i*kmax or q_i*kmin -> exact
// single-pass softmax with one v_exp_f32 per (i,j,attn). Base-2 domain.
// ---------------------------------------------------------------------------
__global__ __launch_bounds__(256) void attn_kernel(
    const float* __restrict__ xp, const float* __restrict__ y,
    const float* __restrict__ pwq, const float* __restrict__ pwk,
    const float* __restrict__ pwv, const float* __restrict__ pbq,
    const float* __restrict__ pbk, const float* __restrict__ pbv,
    const float* __restrict__ pwo, const float* __restrict__ pbo,
    float* __restrict__ out) {
  __shared__ float xs[BDIM * NT];
  __shared__ float ys[BDIM * NT];
  __shared__ float kxmx[NT], kxmn[NT], kymx[NT], kymn[NT];

  const int tid = threadIdx.x;
  const int n0 = blockIdx.x * NT;
  const float wq = *pwq, wk = *pwk, wv = *pwv;
  const float bq = *pbq, bk = *pbk, bv = *pbv;
  const float wo = *pwo, bo = *pbo;
  const float L2E = 1.4426950408889634f;

  // stage the two 128 x NT channel tiles (each n column lives in exactly one block)
  for (int idx = tid; idx < BDIM * NT; idx += 256) {
    int r = idx >> 4;           // row j
    int c = idx & (NT - 1);     // channel
    xs[idx] = xp[(long long)r * NDIM + n0 + c];
    ys[idx] = y [(long long)r * NDIM + n0 + c];
  }
  __syncthreads();

  // per-channel k extremes (exact softmax max via sign of q)
  if (tid < NT) {
    int c = tid;
    float mx = -3.402823466e38f, mn = 3.402823466e38f;
    for (int j = 0; j < BDIM; ++j) {
      float k = xs[j * NT + c] * wk + bk;
      mx = fmaxf(mx, k); mn = fminf(mn, k);
    }
    kxmx[c] = mx; kxmn[c] = mn;
  } else if (tid < 2 * NT) {
    int c = tid - NT;
    float mx = -3.402823466e38f, mn = 3.402823466e38f;
    for (int j = 0; j < BDIM; ++j) {
      float k = ys[j * NT + c] * wk + bk;
      mx = fmaxf(mx, k); mn = fminf(mn, k);
    }
    kymx[c] = mx; kymn[c] = mn;
  }
  __syncthreads();

  for (int p = tid; p < BDIM * NT; p += 256) {   // 8 (i,channel) pairs / thread
    int i = p >> 4;
    int c = p & (NT - 1);
    float xr = xs[i * NT + c];
    float yr = ys[i * NT + c];
    float qx = (xr * wq + bq) * L2E;   // base-2 scaled query
    float qy = (yr * wq + bq) * L2E;
    float m1 = (qx >= 0.f) ? qx * kxmx[c] : qx * kxmn[c];  // self_x : qx vs kx
    float m2 = (qx >= 0.f) ? qx * kymx[c] : qx * kymn[c];  // cross_x: qx vs ky
    float m3 = (qy >= 0.f) ? qy * kymx[c] : qy * kymn[c];  // self_y : qy vs ky
    float m4 = (qy >= 0.f) ? qy * kxmx[c] : qy * kxmn[c];  // cross_y: qy vs kx
    float n1 = 0.f, d1 = 0.f, n2 = 0.f, d2 = 0.f;
    float n3 = 0.f, d3 = 0.f, n4 = 0.f, d4 = 0.f;
    for (int j = 0; j < BDIM; ++j) {
      float rx = xs[j * NT + c];
      float ry = ys[j * NT + c];
      float kxj = rx * wk + bk, vxj = rx * wv + bv;
      float kyj = ry * wk + bk, vyj = ry * wv + bv;
      float e1 = __builtin_amdgcn_exp2f(fmaf(qx, kxj, -m1));
      float e2 = __builtin_amdgcn_exp2f(fmaf(qx, kyj, -m2));
      float e3 = __builtin_amdgcn_exp2f(fmaf(qy, kyj, -m3));
      float e4 = __builtin_amdgcn_exp2f(fmaf(qy, kxj, -m4));
      d1 += e1; n1 = fmaf(e1, vxj, n1);
      d2 += e2; n2 = fmaf(e2, vyj, n2);
      d3 += e3; n3 = fmaf(e3, vyj, n3);
      d4 += e4; n4 = fmaf(e4, vxj, n4);
    }
    float sx = (n1 * __builtin_amdgcn_rcpf(d1) + n2 * __builtin_amdgcn_rcpf(d2)) * wo + 2.0f * bo;
    float sy = (n3 * __builtin_amdgcn_rcpf(d3) + n4 * __builtin_amdgcn_rcpf(d4)) * wo + 2.0f * bo;
    out[(long long)i * (2 * NDIM) + n0 + c]        = xr + sx;  // c_f_x
    out[(long long)i * (2 * NDIM) + NDIM + n0 + c] = yr + sy;  // c_f_y
  }
}

// ---------------------------------------------------------------------------
extern "C" void kernel_launch(void* const* d_in, const int* in_sizes, int n_in,
                              void* d_out, int out_size, void* d_ws, size_t ws_size,
                              hipStream_t stream) {
  (void)in_sizes; (void)n_in; (void)out_size; (void)ws_size;
  const float* x = (const float*)d_in[0];  // [128, 8192, 7, 7]
  const float* y = (const float*)d_in[1];  // [128, 8192]
  float* xp = (float*)d_ws;                // [128, 8192] scratch (4 MB)

  // 1M rows of 49 -> 8192 blocks x 128 rows
  pool_wmma_kernel<<<(BDIM * NDIM) / 128, 256, 0, stream>>>(x, xp);

  attn_kernel<<<NDIM / NT, 256, 0, stream>>>(
      xp, y,
      (const float*)d_in[2], (const float*)d_in[3], (const float*)d_in[4],
      (const float*)d_in[5], (const float*)d_in[6], (const float*)d_in[7],
      (const float*)d_in[8], (const float*)d_in[9],
      (float*)d_out);
}